// sigAct_55783035241195
// MI455X (gfx1250) — compile-verified
//
#include <hip/hip_runtime.h>
#include <math.h>

// ---------------------------------------------------------------------------
// BVP sigmoid-relaxation solver for MI455X (gfx1250).
//   b = sigmoid(10c); lam_b = 100*b
//   u0 = sigmoid(-lam_b * G*(1-2x) / 10)
//   5x: u = 0.5u + 0.5*sigmoid(-((u-x)*10 + lam_b*G*(1-2u)) * 0.1)
// G = 17x17 Gaussian (sigma=2), separable -> two 1D 17-tap passes.
// Each 1D pass is a banded 16x32 @ 32x16 matmul -> 8x V_WMMA_F32_16X16X4_F32.
// ---------------------------------------------------------------------------

typedef float v2f __attribute__((ext_vector_type(2)));
typedef float v8f __attribute__((ext_vector_type(8)));

#define IMG 512
#define TILE 64
#define HALO 8
#define TAPS 17
#define IN_W 80            // TILE + 2*HALO
#define IN_STRIDE 81       // conflict-free (gcd(81,64)=1)
#define HB_STRIDE 65

struct GaussW { float w[TAPS]; };

__device__ __forceinline__ float fast_sigmoid(float z) {
    // 1/(1+e^-z) via v_exp_f32 + v_rcp_f32
    return __builtin_amdgcn_rcpf(1.0f + __expf(-z));
}

__global__ void lamb_kernel(const float* __restrict__ c,
                            float* __restrict__ lamb, int n) {
    int i = blockIdx.x * 256 + threadIdx.x;
    if (i < n) lamb[i] = 100.0f * fast_sigmoid(10.0f * c[i]);
}

template <bool INIT>
__global__ __launch_bounds__(256)
void bvp_iter(const float* __restrict__ src,    // u (or x when INIT)
              const float* __restrict__ x,
              const float* __restrict__ lamb,
              float* __restrict__ dst,
              GaussW gw) {
    __shared__ float in_tile[IN_W * IN_STRIDE];  // conv input s = 1-2u, with halo
    __shared__ float hbuf[IN_W * HB_STRIDE];     // horizontal-pass output (80 rows x 64 cols)
    __shared__ float wsm[TAPS];

    const int tid  = threadIdx.x;
    const int lane = tid & 31;
    const int wave = tid >> 5;
    const int n15  = lane & 15;           // column-within-tile / matrix row id
    const int koff = (lane >> 4) << 1;    // K sub-offset for f32 WMMA A/B layout
    const int mhalf = (lane >> 4) << 3;   // D-matrix row offset per lane half

    const int row0 = blockIdx.y * TILE;
    const int col0 = blockIdx.x * TILE;
    const size_t base = (size_t)blockIdx.z * (IMG * IMG);

    if (tid < TAPS) wsm[tid] = gw.w[tid];

    // Load 80x80 halo tile of s = 1 - 2*src, zero outside the image (SAME pad).
    for (int i = tid; i < IN_W * IN_W; i += 256) {
        int ir = i / IN_W, ic = i - ir * IN_W;
        int gr = row0 - HALO + ir, gc = col0 - HALO + ic;
        float s = 0.0f;
        if (gr >= 0 && gr < IMG && gc >= 0 && gc < IMG)
            s = 1.0f - 2.0f * src[base + (size_t)gr * IMG + gc];
        in_tile[ir * IN_STRIDE + ic] = s;
    }
    __syncthreads();

    // Banded weight operand, per-lane constant for BOTH passes:
    // horizontal B[k][n] = w[k-n] and vertical A[m][k] = w[k-m] share the
    // same per-lane values under the f32 16x4 / 4x16 WMMA layouts:
    //   value[v] at K = 4c + koff + v, row/col index = n15.
    float wband[8][2];
#pragma unroll
    for (int c = 0; c < 8; ++c)
#pragma unroll
        for (int v = 0; v < 2; ++v) {
            int d = 4 * c + koff + v - n15;
            wband[c][v] = (d >= 0 && d < TAPS) ? wsm[d] : 0.0f;
        }

    // ---- Horizontal pass: 20 tiles (5 row-blocks x 4 col-blocks), 8 waves.
    // out[m][n] = sum_k in_tile[R16+m][C16+k] * w[k-n]   (A=input, B=band)
    for (int t = wave; t < 20; t += 8) {
        int R = t >> 2, C = t & 3;
        const float* arow = &in_tile[(R * 16 + n15) * IN_STRIDE + C * 16 + koff];
        v8f acc = {};
#pragma unroll
        for (int c = 0; c < 8; ++c) {
            v2f a; a.x = arow[4 * c]; a.y = arow[4 * c + 1];
            v2f b; b.x = wband[c][0]; b.y = wband[c][1];
            acc = __builtin_amdgcn_wmma_f32_16x16x4_f32(
                false, a, false, b, (short)0, acc, false, false);
        }
#pragma unroll
        for (int v = 0; v < 8; ++v)
            hbuf[(R * 16 + mhalf + v) * HB_STRIDE + C * 16 + n15] = acc[v];
    }
    __syncthreads();

    // ---- Vertical pass + fused update: 16 tiles, 8 waves.
    // out[m][n] = sum_k w[k-m] * hbuf[R16+k][C16+n]   (A=band, B=hbuf)
    const float inv_tau = 10.0f;   // 1/TAU
    const float inv_eps = 0.1f;    // 1/EPSILON
    for (int t = wave; t < 16; t += 8) {
        int R = t >> 2, C = t & 3;
        v8f acc = {};
#pragma unroll
        for (int c = 0; c < 8; ++c) {
            v2f a; a.x = wband[c][0]; a.y = wband[c][1];
            const float* hp = &hbuf[(R * 16 + 4 * c + koff) * HB_STRIDE + C * 16 + n15];
            v2f b; b.x = hp[0]; b.y = hp[HB_STRIDE];
            acc = __builtin_amdgcn_wmma_f32_16x16x4_f32(
                false, a, false, b, (short)0, acc, false, false);
        }
#pragma unroll
        for (int v = 0; v < 8; ++v) {
            int lr = R * 16 + mhalf + v;       // local row in 64x64 tile
            int lc = C * 16 + n15;             // local col
            size_t gidx = base + (size_t)(row0 + lr) * IMG + (col0 + lc);
            float blur = acc[v];
            float lb = lamb[gidx];
            float u;
            if (INIT) {
                u = fast_sigmoid(-(lb * blur) * inv_eps);
            } else {
                // recover u from the LDS conv-input tile: s = 1-2u
                float up = 0.5f * (1.0f - in_tile[(lr + HALO) * IN_STRIDE + (lc + HALO)]);
                float xv = x[gidx];
                float uu = fast_sigmoid(-((up - xv) * inv_tau + lb * blur) * inv_eps);
                u = 0.5f * up + 0.5f * uu;
            }
            dst[gidx] = u;
        }
    }
}

extern "C" void kernel_launch(void* const* d_in, const int* in_sizes, int n_in,
                              void* d_out, int out_size, void* d_ws, size_t ws_size,
                              hipStream_t stream) {
    const float* x = (const float*)d_in[0];
    const float* c = (const float*)d_in[1];
    float* out = (float*)d_out;
    const int n = in_sizes[0];                 // 8*3*512*512
    const int planes = n / (IMG * IMG);        // 24

    float* lamb = (float*)d_ws;
    float* ubuf = lamb + n;

    // Host-side Gaussian taps (sigma=2, normalized 1D; separable == ref 2D).
    GaussW gw;
    {
        float g[TAPS], s = 0.0f;
        for (int t = 0; t < TAPS; ++t) {
            float d = (float)t - 8.0f;
            g[t] = expf(-(d * d) / 8.0f);
            s += g[t];
        }
        for (int t = 0; t < TAPS; ++t) gw.w[t] = g[t] / s;
    }

    lamb_kernel<<<(n + 255) / 256, 256, 0, stream>>>(c, lamb, n);

    dim3 grid(IMG / TILE, IMG / TILE, planes);   // 8 x 8 x 24
    // init + 5 iterations, ping-ponging between ws buffer and d_out;
    // iteration 5 lands in d_out.
    bvp_iter<true ><<<grid, 256, 0, stream>>>(x,    x, lamb, ubuf, gw);
    bvp_iter<false><<<grid, 256, 0, stream>>>(ubuf, x, lamb, out,  gw);
    bvp_iter<false><<<grid, 256, 0, stream>>>(out,  x, lamb, ubuf, gw);
    bvp_iter<false><<<grid, 256, 0, stream>>>(ubuf, x, lamb, out,  gw);
    bvp_iter<false><<<grid, 256, 0, stream>>>(out,  x, lamb, ubuf, gw);
    bvp_iter<false><<<grid, 256, 0, stream>>>(ubuf, x, lamb, out,  gw);
}